// EdgeConv_dgl_67508295958885
// MI455X (gfx1250) — compile-verified
//
#include <hip/hip_runtime.h>

typedef float v2f __attribute__((ext_vector_type(2)));
typedef float v8f __attribute__((ext_vector_type(8)));

#define FEAT 64   // IN_FEAT == OUT_FEAT == 64

__device__ __forceinline__ unsigned enc_f32(float x) {
    unsigned b = __float_as_uint(x);
    return (b & 0x80000000u) ? ~b : (b | 0x80000000u);
}
__device__ __forceinline__ float dec_f32(unsigned u) {
    unsigned b = (u & 0x80000000u) ? (u & 0x7FFFFFFFu) : ~u;
    return __uint_as_float(b);
}

// ---------------- Kernel 0: init encoded-max buffer + degree flags ----------
__global__ void k_init(unsigned* __restrict__ agg, int* __restrict__ deg,
                       int nAgg, int nNodes) {
    int i = blockIdx.x * blockDim.x + threadIdx.x;
    if (i < nAgg)   agg[i] = 0u;      // minimum element of the encoded order
    if (i < nNodes) deg[i] = 0;
}

// ---------------- Kernel 1: fused projection T = feat*Wt^T,
//                  Q = feat*(Wt+Wp)^T + bt + bp  via FP32 WMMA ---------------
__global__ __launch_bounds__(256)
void k_project(const float* __restrict__ feat,
               const float* __restrict__ Wt, const float* __restrict__ bt,
               const float* __restrict__ Wp, const float* __restrict__ bp,
               float* __restrict__ T, float* __restrict__ Q, int nNodes)
{
    const int lane = threadIdx.x & 31;
    const int wave = threadIdx.x >> 5;
    const int rowTiles = (nNodes + 15) >> 4;
    const int tile = blockIdx.x * 8 + wave;           // 8 waves per block
    if (tile >= rowTiles * 4) return;                 // wave-uniform exit
    const int rowTile = tile >> 2;
    const int colTile = tile & 3;

    const int half = lane >> 4;                       // 0: lanes 0-15, 1: 16-31
    const int l15  = lane & 15;
    const int m    = rowTile * 16 + l15;
    const int mCl  = (m < nNodes) ? m : (nNodes - 1); // clamp for ragged tail
    const int colBase = colTile * 16;

    const float* __restrict__ aRow  = feat + (size_t)mCl * FEAT;
    const float* __restrict__ wtRow = Wt + (size_t)(colBase + l15) * FEAT;
    const float* __restrict__ wpRow = Wp + (size_t)(colBase + l15) * FEAT;

    v8f accT = {};
    v8f accP = {};
    // A layout (32-bit, 16x4): VGPR0 holds K = 2*half, VGPR1 holds K = 2*half+1
    // -> each lane loads a contiguous float2 at column kk*4 + 2*half.
    // B (4x16) uses the mirrored K striping, also a contiguous float2 per lane.
    #pragma unroll
    for (int kk = 0; kk < 16; ++kk) {
        const int k0 = kk * 4 + half * 2;
        v2f a  = *(const v2f*)(aRow  + k0);
        v2f bT = *(const v2f*)(wtRow + k0);
        v2f bP = *(const v2f*)(wpRow + k0);
        accT = __builtin_amdgcn_wmma_f32_16x16x4_f32(
                   false, a, false, bT, (short)0, accT, false, false);
        accP = __builtin_amdgcn_wmma_f32_16x16x4_f32(
                   false, a, false, bP, (short)0, accP, false, false);
    }

    const int f = colBase + l15;                      // output feature (N index)
    const float biasQ = bt[f] + bp[f];
    // C/D layout: VGPR r, lanes 0-15 -> M = r ; lanes 16-31 -> M = r + 8
    #pragma unroll
    for (int r = 0; r < 8; ++r) {
        const int node = rowTile * 16 + r + 8 * half;
        if (node < nNodes) {
            const size_t idx = (size_t)node * FEAT + f;
            const float tv = accT[r];
            T[idx] = tv;
            Q[idx] = tv + accP[r] + biasQ;            // theta(dst)+phi(dst)+biases
        }
    }
}

// ---------------- Kernel 2: per-edge message + encoded atomic max -----------
__global__ __launch_bounds__(256)
void k_edges(const int* __restrict__ src, const int* __restrict__ dst,
             const float* __restrict__ T, const float* __restrict__ Q,
             unsigned* __restrict__ agg, int* __restrict__ deg, int nEdges)
{
    const int lane = threadIdx.x & 31;
    int e = blockIdx.x * 8 + (threadIdx.x >> 5);      // one wave per edge
    e = __builtin_amdgcn_readfirstlane(e);            // force scalar index
    if (e >= nEdges) return;
    const int s = src[e];
    const int d = dst[e];

    // lane handles features 2*lane, 2*lane+1 (fully coalesced float2)
    const v2f t = *(const v2f*)(T + (size_t)s * FEAT + 2 * lane);
    const v2f q = *(const v2f*)(Q + (size_t)d * FEAT + 2 * lane);

    unsigned* p = agg + (size_t)d * FEAT + 2 * lane;
    atomicMax(p,     enc_f32(q.x - t.x));
    atomicMax(p + 1, enc_f32(q.y - t.y));
    if (lane == 0) deg[d] = 1;                        // idempotent flag
}

// ---------------- Kernel 3: decode + zero-fill zero-degree nodes ------------
__global__ void k_finalize(const int* __restrict__ deg, float* __restrict__ out,
                           int total)
{
    const int i = blockIdx.x * blockDim.x + threadIdx.x;
    if (i >= total) return;
    const unsigned u = ((const unsigned*)out)[i];
    out[i] = (deg[i >> 6] != 0) ? dec_f32(u) : 0.0f;
}

extern "C" void kernel_launch(void* const* d_in, const int* in_sizes, int n_in,
                              void* d_out, int out_size, void* d_ws, size_t ws_size,
                              hipStream_t stream)
{
    const float* feat = (const float*)d_in[0];
    const int*   src  = (const int*)  d_in[1];
    const int*   dst  = (const int*)  d_in[2];
    const float* Wt   = (const float*)d_in[3];
    const float* bt   = (const float*)d_in[4];
    const float* Wp   = (const float*)d_in[5];
    const float* bp   = (const float*)d_in[6];

    const int nNodes = in_sizes[0] / FEAT;
    const int nEdges = in_sizes[1];
    const int nAgg   = nNodes * FEAT;

    // workspace layout: T [N*64 f32] | Q [N*64 f32] | deg [N i32]
    float* T   = (float*)d_ws;
    float* Q   = T + (size_t)nNodes * FEAT;
    int*   deg = (int*)(Q + (size_t)nNodes * FEAT);
    unsigned* agg = (unsigned*)d_out;                 // encoded max lives in d_out

    k_init<<<(nAgg + 255) / 256, 256, 0, stream>>>(agg, deg, nAgg, nNodes);

    const int rowTiles = (nNodes + 15) / 16;
    const int tiles    = rowTiles * 4;                // 4 col tiles (64 feats)
    k_project<<<(tiles + 7) / 8, 256, 0, stream>>>(feat, Wt, bt, Wp, bp,
                                                   T, Q, nNodes);

    k_edges<<<(nEdges + 7) / 8, 256, 0, stream>>>(src, dst, T, Q, agg, deg,
                                                  nEdges);

    k_finalize<<<(nAgg + 255) / 256, 256, 0, stream>>>(deg, (float*)d_out, nAgg);
}